// MultiGCN_relation_55310588838066
// MI455X (gfx1250) — compile-verified
//
#include <hip/hip_runtime.h>
#include <math.h>

// ---------------------------------------------------------------------------
// MultiGCN relation network, MI455X (gfx1250, wave32).
// N=512, D=512, H=64, OUT=32.  All f32; GEMMs use V_WMMA_F32_16X16X4_F32.
// ---------------------------------------------------------------------------

#define N    512
#define DIM  512
#define H    64
#define OUT  32

typedef __attribute__((ext_vector_type(2))) float v2f;
typedef __attribute__((ext_vector_type(8))) float v8f;

// One wave computes one 16x16 f32 tile: D = A(16xK) * B(Kx16).
//   Arow : pointer to A + m0*lda               (row-major, stride lda)
//   Bcol : pointer to B + (n0 + lane%16)       (row-major, stride ldb) -- per-lane column base
// A-frag layout (16x4, 32-bit): lanes 0-15 hold K,K+1; lanes 16-31 hold K+2,K+3 (VGPR0/1).
// B-frag layout (4x16): VGPR0 = rows {K (lanes 0-15), K+2 (lanes 16-31)}, VGPR1 = rows {K+1, K+3}.
__device__ inline v8f gemm16x16_f32(const float* __restrict__ Arow, int lda,
                                    const float* __restrict__ Bcol, int ldb,
                                    int K, int lane) {
  v8f c = {};
  const float* ap = Arow + (lane & 15) * lda + ((lane >> 4) << 1);
  const float* bp = Bcol + ((lane >> 4) << 1) * ldb;
  for (int k = 0; k < K; k += 4) {
    v2f a; a.x = ap[0];  a.y = ap[1];
    v2f b; b.x = bp[0];  b.y = bp[ldb];
    c = __builtin_amdgcn_wmma_f32_16x16x4_f32(false, a, false, b, (short)0, c,
                                              false, false);
    ap += 4;
    bp += 4 * ldb;
  }
  return c;
}

// C/D 16x16 f32 layout: VGPR v -> row m0+v (lanes 0-15), row m0+v+8 (lanes 16-31).
__device__ inline void store_tile16x16(float* __restrict__ D, int ldd,
                                       int m0, int n0, v8f c, int lane) {
  int col   = n0 + (lane & 15);
  int rbase = m0 + ((lane >> 4) << 3);
#pragma unroll
  for (int v = 0; v < 8; ++v) D[(rbase + v) * ldd + col] = c[v];
}

// ---------------------------------------------------------------------------
// Kernel 1: PQ = F @ [W1_top | W1_bot]   -> (512 x 128)
//   PQ[:,0:64]  = F @ rel_w1[0:512,:]    (P, the "j" half)
//   PQ[:,64:128]= F @ rel_w1[512:1024,:] (Q, the "i" half)
// 256 tiles of 16x16; 8 waves per block -> grid 32.
// ---------------------------------------------------------------------------
__global__ void gemm_pq_kernel(const float* __restrict__ F,
                               const float* __restrict__ w1,
                               float* __restrict__ PQ) {
  const int wave = threadIdx.x >> 5;
  const int lane = threadIdx.x & 31;
  const int tile = blockIdx.x * 8 + wave;   // 0..255
  const int m0 = (tile >> 3) * 16;          // 32 row tiles
  const int n0 = (tile & 7) * 16;           // 8 col tiles
  const int c  = n0 + (lane & 15);          // 0..127
  const float* Bcol = (c < H) ? (w1 + c) : (w1 + DIM * H + (c - H));
  v8f acc = gemm16x16_f32(F + m0 * DIM, DIM, Bcol, H, DIM, lane);
  store_tile16x16(PQ, 2 * H, m0, n0, acc, lane);
}

// ---------------------------------------------------------------------------
// Kernel 2: XW = F @ gcn_w  -> (512 x 32).  64 tiles -> grid 8.
// ---------------------------------------------------------------------------
__global__ void gemm_xw_kernel(const float* __restrict__ F,
                               const float* __restrict__ W,
                               float* __restrict__ XW) {
  const int wave = threadIdx.x >> 5;
  const int lane = threadIdx.x & 31;
  const int tile = blockIdx.x * 8 + wave;   // 0..63
  const int m0 = (tile >> 1) * 16;
  const int n0 = (tile & 1) * 16;
  const float* Bcol = W + n0 + (lane & 15);
  v8f acc = gemm16x16_f32(F + m0 * DIM, DIM, Bcol, OUT, DIM, lane);
  store_tile16x16(XW, OUT, m0, n0, acc, lane);
}

// ---------------------------------------------------------------------------
// Kernel 3: adj[i,j] = sigmoid( sum_h relu(P[j,h]+Q[i,h]+b1[h]) * w2[h] + b2 )
// 16x16 output tile per block, P/Q rows staged in LDS.
// ---------------------------------------------------------------------------
__global__ void relnet_pair_kernel(const float* __restrict__ PQ,
                                   const float* __restrict__ b1,
                                   const float* __restrict__ w2,
                                   const float* __restrict__ b2p,
                                   float* __restrict__ adj) {
  __shared__ float Pj[16][H];
  __shared__ float Qi[16][H];
  __shared__ float b1s[H];
  __shared__ float w2s[H];
  const int tid = threadIdx.x;
  const int j0 = blockIdx.x * 16;
  const int i0 = blockIdx.y * 16;

  for (int t = tid; t < 16 * H; t += 256) {
    int r = t >> 6, h = t & (H - 1);
    Pj[r][h] = PQ[(j0 + r) * (2 * H) + h];
    Qi[r][h] = PQ[(i0 + r) * (2 * H) + H + h];
  }
  if (tid < H) { b1s[tid] = b1[tid]; w2s[tid] = w2[tid]; }
  __syncthreads();

  const int tj = tid & 15;
  const int ti = tid >> 4;
  float acc = 0.0f;
#pragma unroll 8
  for (int h = 0; h < H; ++h) {
    float t = Pj[tj][h] + Qi[ti][h] + b1s[h];
    acc = fmaf(fmaxf(t, 0.0f), w2s[h], acc);
  }
  float z = acc + b2p[0];
  adj[(i0 + ti) * N + (j0 + tj)] = 1.0f / (1.0f + expf(-z));
}

// ---------------------------------------------------------------------------
// Kernel 4: dvec[i] = sqrt( 2 + sum_j adj[i,j] )   (rowsum(I+adj)+1)
// ---------------------------------------------------------------------------
__global__ void dvec_kernel(const float* __restrict__ adj,
                            float* __restrict__ dvec) {
  __shared__ float red[256];
  const int i = blockIdx.x, tid = threadIdx.x;
  red[tid] = adj[i * N + tid] + adj[i * N + tid + 256];
  __syncthreads();
  for (int off = 128; off > 0; off >>= 1) {
    if (tid < off) red[tid] += red[tid + off];
    __syncthreads();
  }
  if (tid == 0) dvec[i] = sqrtf(red[0] + 2.0f);
}

// ---------------------------------------------------------------------------
// Kernel 5: A_out[i,j] = ((i==j) + adj[i,j]) / (dvec[i]*dvec[j])
// ---------------------------------------------------------------------------
__global__ void norm_kernel(const float* __restrict__ adj,
                            const float* __restrict__ dvec,
                            float* __restrict__ Aout) {
  const int idx = blockIdx.x * 256 + threadIdx.x;
  const int i = idx >> 9;
  const int j = idx & (N - 1);
  float a = adj[idx] + (i == j ? 1.0f : 0.0f);
  Aout[idx] = a / (dvec[i] * dvec[j]);
}

// ---------------------------------------------------------------------------
// Kernel 6: x = relu( bn_gamma/sqrt(1+eps) * (A @ XW + gcn_b) + bn_beta )
// ---------------------------------------------------------------------------
__global__ void gemm_out_kernel(const float* __restrict__ Anorm,
                                const float* __restrict__ XW,
                                const float* __restrict__ gcn_b,
                                const float* __restrict__ gamma,
                                const float* __restrict__ beta,
                                float* __restrict__ X) {
  const int wave = threadIdx.x >> 5;
  const int lane = threadIdx.x & 31;
  const int tile = blockIdx.x * 8 + wave;   // 0..63
  const int m0 = (tile >> 1) * 16;
  const int n0 = (tile & 1) * 16;
  const float* Bcol = XW + n0 + (lane & 15);
  v8f acc = gemm16x16_f32(Anorm + m0 * N, N, Bcol, OUT, N, lane);

  const int col = n0 + (lane & 15);
  const float inv_std = 0.99999500003749969f;  // 1/sqrt(1 + 1e-5)
  const float sc = gamma[col] * inv_std;
  const float bt = beta[col];
  const float gb = gcn_b[col];
  const int rbase = m0 + ((lane >> 4) << 3);
#pragma unroll
  for (int v = 0; v < 8; ++v) {
    float t = (acc[v] + gb) * sc + bt;
    X[(rbase + v) * OUT + col] = fmaxf(t, 0.0f);
  }
}

// ---------------------------------------------------------------------------
// Launch
// ---------------------------------------------------------------------------
extern "C" void kernel_launch(void* const* d_in, const int* in_sizes, int n_in,
                              void* d_out, int out_size, void* d_ws, size_t ws_size,
                              hipStream_t stream) {
  const float* F     = (const float*)d_in[0];  // (512,512)
  const float* w1    = (const float*)d_in[1];  // (1024,64)
  const float* b1    = (const float*)d_in[2];  // (64,)
  const float* w2    = (const float*)d_in[3];  // (64,1)
  const float* b2    = (const float*)d_in[4];  // (1,)
  const float* gw    = (const float*)d_in[5];  // (512,32)
  const float* gb    = (const float*)d_in[6];  // (32,)
  const float* gamma = (const float*)d_in[7];  // (32,)
  const float* beta  = (const float*)d_in[8];  // (32,)

  float* out_x = (float*)d_out;                // (512,32) flat
  float* out_A = out_x + N * OUT;              // (512,512) flat

  float* ws   = (float*)d_ws;
  float* PQ   = ws;                            // 512*128
  float* adj  = PQ + N * 2 * H;                // 512*512
  float* XW   = adj + N * N;                   // 512*32
  float* dvec = XW + N * OUT;                  // 512

  gemm_pq_kernel   <<<32,            256, 0, stream>>>(F, w1, PQ);
  gemm_xw_kernel   <<<8,             256, 0, stream>>>(F, gw, XW);
  relnet_pair_kernel<<<dim3(32, 32), 256, 0, stream>>>(PQ, b1, w2, b2, adj);
  dvec_kernel      <<<N,             256, 0, stream>>>(adj, dvec);
  norm_kernel      <<<(N * N) / 256, 256, 0, stream>>>(adj, dvec, out_A);
  gemm_out_kernel  <<<8,             256, 0, stream>>>(out_A, XW, gb, gamma, beta, out_x);
}